// MADFNet_45079976739073
// MI455X (gfx1250) — compile-verified
//
#include <hip/hip_runtime.h>
#include <hip/hip_bf16.h>

typedef __attribute__((ext_vector_type(2))) float v2f;
typedef __attribute__((ext_vector_type(8))) float v8f;

#define NT 4   // N-tiles (of 16 cols) per wave: 16x64 output strip

// ---------------------------------------------------------------------------
// Tap sweep for one 4-channel chunk: KK*KK taps, NT wmma per tap.
// TAIL=false: all 4 chunk channels valid (no per-element selects on B).
// A gather is branchless: clamped address + cndmask on the halo predicate.
// ---------------------------------------------------------------------------
template <int KK, int NTT, bool TAIL>
__device__ __forceinline__ void conv_taps(
    v8f (&acc)[NTT], const float* const (&pA)[2], const bool (&cv)[2],
    const float* const (&wrow)[NTT], const int (&wOff)[2],
    int by, int bx, int Hin, int Win, int plane, bool mvalid)
{
    for (int ky = 0; ky < KK; ++ky) {
        const int iy = by + ky;
        const bool yok = mvalid & (iy >= 0) & (iy < Hin);
        const int rowoff = iy * Win;
#pragma unroll
        for (int kx = 0; kx < KK; ++kx) {
            const int ix = bx + kx;
            const bool pok = yok & (ix >= 0) & (ix < Win);
            int soff = rowoff + ix;
            soff = soff < 0 ? 0 : (soff > plane - 1 ? plane - 1 : soff);
            const int koff = ky * KK + kx;
            v2f a;
#pragma unroll
            for (int t = 0; t < 2; ++t) {
                const float v = pA[t][soff];          // always in-bounds (clamped)
                const bool ok = TAIL ? (pok && cv[t]) : pok;
                a[t] = ok ? v : 0.0f;
            }
#pragma unroll
            for (int n = 0; n < NTT; ++n) {
                v2f b;
#pragma unroll
                for (int t = 0; t < 2; ++t) {
                    const float bv = wrow[n][wOff[t] + koff]; // imm offset (kx unrolled)
                    b[t] = (TAIL && !cv[t]) ? 0.0f : bv;
                }
                acc[n] = __builtin_amdgcn_wmma_f32_16x16x4_f32(
                    false, a, false, b, (short)0, acc[n], false, false);
            }
        }
    }
}

// ---------------------------------------------------------------------------
// Implicit-GEMM conv (NCHW, N=1) on V_WMMA_F32_16X16X4_F32.
// K enumerated as (ic, ky, kx); each wmma consumes 4 channels at one tap.
// Channel-chunk setup (pointer select, 64-bit math, tail predicate) hoisted
// out of the tap loops; the <=1 tail chunk runs AFTER the main loop so the
// loop body has a single conv_taps call site (no accumulator phi copies).
// One wave computes a 16(M) x 64(N) output strip.
// HAS2 fuses channel-concat (skip ++ deconv) into the A gather.
// ---------------------------------------------------------------------------
template <int KK, bool HAS2>
__global__ void __launch_bounds__(32) conv2d_wmma(
    const float* __restrict__ in1, int C1,
    const float* __restrict__ in2, int C2,
    const float* __restrict__ w,   // (Cout, Cin*KK*KK) row-major (OIHW flat)
    const float* __restrict__ bias,
    float* __restrict__ out,       // (Cout, Hout*Wout)
    int Hin, int Win, int Cout, int s, int pad,
    int Hout, int Wout, int act)   // act: 0 = none, 1 = relu
{
    constexpr int KK2 = KK * KK;
    const int lane  = threadIdx.x;
    const int half  = lane >> 4;     // 0: K slots {0,1}, 1: K slots {2,3}
    const int l15   = lane & 15;
    const int tileM  = blockIdx.x;
    const int tileN0 = blockIdx.y * NT;

    const int M   = Hout * Wout;
    const int Cin = HAS2 ? (C1 + C2) : C1;

    const int mA = tileM * 16 + l15;
    const bool mvalid = (mA < M);
    int oy = 0, ox = 0;
    if (mvalid) { oy = mA / Wout; ox = mA - oy * Wout; }
    const int by = oy * s - pad;
    const int bx = ox * s - pad;
    const int plane = Hin * Win;

    int  nB[NT];
    bool nv[NT];
    const float* wrow[NT];
    v8f acc[NT];
#pragma unroll
    for (int n = 0; n < NT; ++n) {
        nB[n] = (tileN0 + n) * 16 + l15;
        nv[n] = (nB[n] < Cout);
        wrow[n] = w + (size_t)(nv[n] ? nB[n] : 0) * (size_t)(Cin * KK2);
        const float bv = nv[n] ? bias[nB[n]] : 0.0f;
#pragma unroll
        for (int r = 0; r < 8; ++r) acc[n][r] = bv;
    }

    const int icBase  = 2 * half;    // this lane supplies channels ic0+icBase, +1
    const int CinMain = Cin & ~3;

    // main chunks: all 4 channels valid, single call site (no acc phi copies)
    for (int ic0 = 0; ic0 < CinMain; ic0 += 4) {
        const float* pA[2];
        bool cv[2];
        int  wOff[2];
#pragma unroll
        for (int t = 0; t < 2; ++t) {
            const int ict = ic0 + icBase + t;
            cv[t]   = true;
            wOff[t] = ict * KK2;
            if (HAS2 && ict >= C1) pA[t] = in2 + (size_t)(ict - C1) * plane;
            else                   pA[t] = in1 + (size_t)ict * plane;
        }
        conv_taps<KK, NT, false>(acc, pA, cv, wrow, wOff, by, bx, Hin, Win, plane, mvalid);
    }

    // at most one tail chunk (Cin % 4 != 0): per-slot validity predicates
    if (CinMain < Cin) {
        const float* pA[2];
        bool cv[2];
        int  wOff[2];
#pragma unroll
        for (int t = 0; t < 2; ++t) {
            const int ict = CinMain + icBase + t;
            const int icc = (ict < Cin) ? ict : (Cin - 1);
            cv[t]   = (ict < Cin);
            wOff[t] = icc * KK2;
            if (HAS2 && icc >= C1) pA[t] = in2 + (size_t)(icc - C1) * plane;
            else                   pA[t] = in1 + (size_t)icc * plane;
        }
        conv_taps<KK, NT, true>(acc, pA, cv, wrow, wOff, by, bx, Hin, Win, plane, mvalid);
    }

    // D layout: lanes 0-15 rows 0..7 (vgpr r -> M=r), lanes 16-31 rows 8..15
#pragma unroll
    for (int n = 0; n < NT; ++n) {
#pragma unroll
        for (int r = 0; r < 8; ++r) {
            const int m = tileM * 16 + half * 8 + r;
            if (m < M && nv[n]) {
                float v = acc[n][r];
                if (act == 1) v = fmaxf(v, 0.0f);
                out[(size_t)nB[n] * M + m] = v;
            }
        }
    }
}

// ---------------------------------------------------------------------------
// Per-pixel dynamic conv with fused filter generation:
//   cr[o,l] = relu( sum_k patch[l,k] * ( b1[k*Cout+o] + sum_c w1[k*Cout+o][c]*hid[c,l] ) )
// ---------------------------------------------------------------------------
__global__ void __launch_bounds__(128) dynconv_kernel(
    const float* __restrict__ pre,   // (Cin, Hin, Win)
    const float* __restrict__ hid,   // (16, Hout*Wout)
    const float* __restrict__ w1,    // (Cin*k*k*Cout, 16)
    const float* __restrict__ b1,    // (Cin*k*k*Cout)
    float* __restrict__ out,         // (Cout, Hout*Wout)
    int Cin, int Hin, int Win, int Cout, int kk, int s, int pad,
    int Hout, int Wout)
{
    extern __shared__ float sm[];    // [K] patch, [16] hidden
    const int kk2 = kk * kk;
    const int K   = Cin * kk2;
    const int l   = blockIdx.x;
    const int M   = Hout * Wout;
    const int oy  = l / Wout, ox = l - oy * Wout;
    const int by  = oy * s - pad, bx = ox * s - pad;

    float* sp = sm;
    float* sh = sm + K;
    for (int k = threadIdx.x; k < K; k += blockDim.x) {
        const int ic = k / kk2, rem = k - ic * kk2;
        const int ky = rem / kk, kx = rem - ky * kk;
        const int iy = by + ky, ix = bx + kx;
        sp[k] = (iy >= 0 && iy < Hin && ix >= 0 && ix < Win)
                    ? pre[(size_t)ic * Hin * Win + iy * Win + ix] : 0.0f;
    }
    if (threadIdx.x < 16) sh[threadIdx.x] = hid[(size_t)threadIdx.x * M + l];
    __syncthreads();

    float h[16];
#pragma unroll
    for (int c = 0; c < 16; ++c) h[c] = sh[c];

    for (int o = threadIdx.x; o < Cout; o += blockDim.x) {
        float acc = 0.0f;
        for (int k = 0; k < K; ++k) {
            const int row = k * Cout + o;
            const float* wr = w1 + ((size_t)row << 4);
            __builtin_prefetch(wr + ((size_t)Cout << 4), 0, 1);
            float fv = b1[row];
#pragma unroll
            for (int c = 0; c < 16; c += 4) {
                const float4 w4 = *(const float4*)(wr + c);
                fv += w4.x * h[c] + w4.y * h[c + 1] + w4.z * h[c + 2] + w4.w * h[c + 3];
            }
            acc += sp[k] * fv;
        }
        out[(size_t)o * M + l] = fmaxf(acc, 0.0f);
    }
}

// ---------------------------------------------------------------------------
// ConvTranspose2d k=4, s=2, p=1.  w layout: (Ci, Co, 4, 4).
// ---------------------------------------------------------------------------
__global__ void deconv_kernel(
    const float* __restrict__ in, const float* __restrict__ w,
    const float* __restrict__ bias, float* __restrict__ out,
    int Ci, int Co, int H, int W)
{
    const int Ho = 2 * H, Wo = 2 * W;
    const int idx = blockIdx.x * blockDim.x + threadIdx.x;
    const int total = Co * Ho * Wo;
    if (idx >= total) return;
    const int oc  = idx / (Ho * Wo);
    const int rem = idx - oc * Ho * Wo;
    const int oy  = rem / Wo, ox = rem - oy * Wo;

    float acc = bias[oc];
    const int ky0 = (oy + 1) & 1;
    const int kx0 = (ox + 1) & 1;
    for (int ky = ky0; ky < 4; ky += 2) {
        const int iy = (oy + 1 - ky) >> 1;
        if (iy < 0 || iy >= H) continue;
        for (int kx = kx0; kx < 4; kx += 2) {
            const int ix = (ox + 1 - kx) >> 1;
            if (ix < 0 || ix >= W) continue;
            const float* ip = in + (size_t)iy * W + ix;
            const float* wp = w + (size_t)oc * 16 + ky * 4 + kx;
            for (int ic = 0; ic < Ci; ++ic)
                acc += ip[(size_t)ic * H * W] * wp[(size_t)ic * Co * 16];
        }
    }
    out[idx] = acc;
}

// ---------------------------------------------------------------------------
// Per-channel BatchNorm (biased var, eps=1e-5) + LeakyReLU(0.2), in place.
// ---------------------------------------------------------------------------
__global__ void __launch_bounds__(256) bn_leaky_kernel(
    float* __restrict__ x, const float* __restrict__ g,
    const float* __restrict__ beta, int M)
{
    __shared__ float s1[256], s2[256];
    const int c = blockIdx.x;
    float* p = x + (size_t)c * M;
    float sum = 0.0f, sq = 0.0f;
    for (int i = threadIdx.x; i < M; i += 256) {
        const float v = p[i];
        sum += v; sq += v * v;
    }
    s1[threadIdx.x] = sum; s2[threadIdx.x] = sq;
    __syncthreads();
    for (int off = 128; off > 0; off >>= 1) {
        if (threadIdx.x < off) {
            s1[threadIdx.x] += s1[threadIdx.x + off];
            s2[threadIdx.x] += s2[threadIdx.x + off];
        }
        __syncthreads();
    }
    const float mean = s1[0] / (float)M;
    const float var  = s2[0] / (float)M - mean * mean;
    const float sc   = g[c] * rsqrtf(var + 1e-5f);
    const float bb   = beta[c];
    for (int i = threadIdx.x; i < M; i += 256) {
        const float v = (p[i] - mean) * sc + bb;
        p[i] = (v < 0.0f) ? 0.2f * v : v;
    }
}

__global__ void concat2_kernel(const float* __restrict__ a, int na,
                               const float* __restrict__ b, int nb,
                               float* __restrict__ out)
{
    const int i = blockIdx.x * blockDim.x + threadIdx.x;
    if (i < na) out[i] = a[i];
    else if (i < na + nb) out[i] = b[i - na];
}

// ---------------------------------------------------------------------------
// Host orchestration
// ---------------------------------------------------------------------------
static inline void launch_conv(const float* in1, int C1, const float* in2, int C2,
                               const float* w, const float* b, float* out,
                               int Hin, int Win, int Cout, int k, int s, int pad,
                               int Hout, int Wout, int act, hipStream_t st)
{
    dim3 grid((Hout * Wout + 15) / 16, (Cout + 16 * NT - 1) / (16 * NT));
#define LC(KKV, H2) conv2d_wmma<KKV, H2><<<grid, 32, 0, st>>>( \
        in1, C1, in2, C2, w, b, out, Hin, Win, Cout, s, pad, Hout, Wout, act)
    if (C2 > 0) {
        switch (k) {
            case 1: LC(1, true); break;
            case 3: LC(3, true); break;
            case 5: LC(5, true); break;
            default: LC(7, true); break;
        }
    } else {
        switch (k) {
            case 1: LC(1, false); break;
            case 3: LC(3, false); break;
            case 5: LC(5, false); break;
            default: LC(7, false); break;
        }
    }
#undef LC
}

extern "C" void kernel_launch(void* const* d_in, const int* in_sizes, int n_in,
                              void* d_out, int out_size, void* d_ws, size_t ws_size,
                              hipStream_t stream)
{
    if (n_in < 84) return;
    static const int SP_cin[8]  = {0, 6, 16, 32, 64, 128, 128, 128};
    static const int SP_cout[8] = {0, 16, 32, 64, 128, 128, 128, 128};
    static const int SP_k[8]    = {0, 7, 5, 3, 3, 3, 3, 3};
    static const int SP_p[8]    = {0, 3, 2, 1, 1, 1, 1, 1};
    static const int UP_co[8]   = {0, 64, 128, 256, 512, 512, 512, 512};
    static const int DEC_ci[8]  = {0, 64, 128, 256, 512, 512, 512, 512};
    static const int DEC_co[8]  = {0, 3, 64, 128, 256, 512, 512, 512};
    static const int ATT_co[8]  = {0, 3, 64, 128, 256, 512, 512, 512};

    const float* INP  = (const float*)d_in[0];
    const float* MASK = (const float*)d_in[1];
    // params flattened in jax-pytree (sorted key) order
    const float *att_b[8], *att_w[8];
    for (int i = 1; i <= 7; ++i) { att_b[i] = (const float*)d_in[2 + 2*(i-1)];
                                   att_w[i] = (const float*)d_in[3 + 2*(i-1)]; }
    const float *dec_b[8], *dec_beta[8], *dec_g[8], *dec_w[8];
    dec_b[1] = (const float*)d_in[16]; dec_w[1] = (const float*)d_in[17];
    dec_beta[1] = nullptr; dec_g[1] = nullptr;
    for (int i = 2; i <= 7; ++i) {
        const int b0 = 18 + 4 * (i - 2);
        dec_b[i]    = (const float*)d_in[b0];
        dec_beta[i] = (const float*)d_in[b0 + 1];
        dec_g[i]    = (const float*)d_in[b0 + 2];
        dec_w[i]    = (const float*)d_in[b0 + 3];
    }
    const float *fg_b0[8], *fg_b1[8], *fg_w0[8], *fg_w1[8];
    for (int i = 1; i <= 7; ++i) {
        const int b0 = 42 + 4 * (i - 1);
        fg_b0[i] = (const float*)d_in[b0];
        fg_b1[i] = (const float*)d_in[b0 + 1];
        fg_w0[i] = (const float*)d_in[b0 + 2];
        fg_w1[i] = (const float*)d_in[b0 + 3];
    }
    const float *up_b[8], *up_w[8];
    for (int i = 1; i <= 7; ++i) { up_b[i] = (const float*)d_in[70 + 2*(i-1)];
                                   up_w[i] = (const float*)d_in[71 + 2*(i-1)]; }

    // workspace carve (floats)
    float* ws = (float*)d_ws;
    size_t off = 0;
    auto alloc = [&](size_t n) { float* p = ws + off; off += n; return p; };
    float* h0 = alloc((size_t)6 * 65536);
    float *hid[8], *cr[8], *hb[8], *db[8], *hatt[8];
    for (int i = 1; i <= 7; ++i) {
        const int S = 256 >> i; const size_t M = (size_t)S * S;
        hid[i] = alloc(16 * M);
        cr[i]  = alloc((size_t)SP_cout[i] * M);
        hb[i]  = alloc((size_t)UP_co[i] * M);
    }
    for (int i = 1; i <= 7; ++i) {
        const int So = 256 >> (i - 1);
        db[i] = alloc((size_t)DEC_co[i] * So * So);
    }
    for (int i = 2; i <= 7; ++i) {
        const int So = 256 >> (i - 1);
        hatt[i] = alloc((size_t)ATT_co[i] * So * So);
    }
    if (off * sizeof(float) > ws_size) return;

    // h_0 = concat(input, mask)
    {
        const int n = 6 * 65536;
        concat2_kernel<<<(n + 255) / 256, 256, 0, stream>>>(INP, 3 * 65536, MASK, 3 * 65536, h0);
    }

    // ---- encoder ----
    const float* m_pre = MASK;
    const float* pre   = h0;
    for (int i = 1; i <= 7; ++i) {
        const int Sp = 256 >> (i - 1), S = 256 >> i, M = S * S;
        const int k = SP_k[i], p = SP_p[i];
        launch_conv(m_pre, (i == 1) ? 3 : 16, nullptr, 0, fg_w0[i], fg_b0[i],
                    hid[i], Sp, Sp, 16, k, 2, p, S, S, 1, stream);
        const int K = SP_cin[i] * k * k;
        dynconv_kernel<<<M, 128, (size_t)(K + 16) * sizeof(float), stream>>>(
            pre, hid[i], fg_w1[i], fg_b1[i], cr[i],
            SP_cin[i], Sp, Sp, SP_cout[i], k, 2, p, S, S);
        launch_conv(cr[i], SP_cout[i], nullptr, 0, up_w[i], up_b[i],
                    hb[i], S, S, UP_co[i], 1, 1, 0, S, S, 1, stream);
        m_pre = hid[i];
        pre   = cr[i];
    }

    // ---- decoder ----
    const float* cur = hb[7];
    for (int i = 7; i >= 1; --i) {
        const int S = 256 >> i, So = 2 * S;
        const int total = DEC_co[i] * So * So;
        deconv_kernel<<<(total + 255) / 256, 256, 0, stream>>>(
            cur, dec_w[i], dec_b[i], db[i], DEC_ci[i], DEC_co[i], S, S);
        if (i != 1)
            bn_leaky_kernel<<<DEC_co[i], 256, 0, stream>>>(db[i], dec_g[i], dec_beta[i], So * So);
        const float* skip = (i == 1) ? h0 : hb[i - 1];
        const int skipC   = (i == 1) ? 6 : UP_co[i - 1];
        float* outp       = (i == 1) ? (float*)d_out : hatt[i];
        launch_conv(skip, skipC, db[i], DEC_co[i], att_w[i], att_b[i], outp,
                    So, So, ATT_co[i], 3, 1, 1, So, So, (i == 1) ? 0 : 1, stream);
        cur = outp;
    }
    (void)in_sizes; (void)out_size;
}